// LabelEncoder_17205638987990
// MI455X (gfx1250) — compile-verified
//
#include <hip/hip_runtime.h>
#include <hip/hip_bf16.h>

// ---------------- problem constants ----------------
#define BB 16
#define AA 8400
#define GG 64
#define CC 80
#define KK 10
#define EPSF 1e-9f
#define ATILES 525   // AA/16
#define GTILES 4     // GG/16
#define NTILES (BB*GTILES*ATILES)   // 33600

typedef float v2f __attribute__((ext_vector_type(2)));
typedef float v8f __attribute__((ext_vector_type(8)));

// ---------------- CIoU (atan terms precomputed) ----------------
__device__ __forceinline__ float ciou_f(float4 g, float4 d, float atg, float atd) {
    float x1 = fmaxf(g.x, d.x), y1 = fmaxf(g.y, d.y);
    float x2 = fminf(g.z, d.z), y2 = fminf(g.w, d.w);
    float inter = fmaxf(x2 - x1, 0.0f) * fmaxf(y2 - y1, 0.0f);
    float w1 = g.z - g.x, h1 = g.w - g.y;
    float w2 = d.z - d.x, h2 = d.w - d.y;
    float uni = w1 * h1 + w2 * h2 - inter;
    float iou = inter / (uni + EPSF);
    float cw = fmaxf(g.z, d.z) - fminf(g.x, d.x);
    float ch = fmaxf(g.w, d.w) - fminf(g.y, d.y);
    float c2 = cw * cw + ch * ch + EPSF;
    float dx = g.x + g.z - d.x - d.z;
    float dy = g.y + g.w - d.y - d.w;
    float rho2 = (dx * dx + dy * dy) * 0.25f;
    float dv = atg - atd;
    const float k = 4.0f / (3.14159265358979323846f * 3.14159265358979323846f);
    float v = k * dv * dv;
    float aa = v / (v - iou + 1.0f + EPSF);
    return iou - (rho2 / c2 + v * aa);
}

// ---------------- 0: zero atomic buffers ----------------
__global__ void tal_init_kernel(unsigned long long* __restrict__ best,
                                float* __restrict__ norm) {
    int i = blockIdx.x * blockDim.x + threadIdx.x;
    if (i < BB * AA) { best[i] = 0ull; norm[i] = 0.0f; }
}

// ---------------- 1: precompute arctan(w/(h+eps)) ----------------
__global__ void tal_atan_kernel(const float* __restrict__ dec,
                                const float* __restrict__ gtb,
                                float* __restrict__ atan_dec,
                                float* __restrict__ atan_gt) {
    int i = blockIdx.x * blockDim.x + threadIdx.x;
    if (i < BB * AA) {
        float4 d = *(const float4*)(dec + (size_t)i * 4);
        atan_dec[i] = atanf((d.z - d.x) / (d.w - d.y + EPSF));
    }
    int j = i - BB * AA;
    if (j >= 0 && j < BB * GG) {
        float4 g = *(const float4*)(gtb + (size_t)j * 4);
        atan_gt[j] = atanf((g.z - g.x) / (g.w - g.y + EPSF));
    }
}

// ---------------- 2: align metric via exact f32 WMMA gather ----------------
// bbox_scores[G,A] = onehot(labels)[G,C] x scores^T[C,A], tiled 16x16,
// K-steps of 4 over C=80.  One tile per wave, 8 waves per block.
__global__ __launch_bounds__(256) void tal_wmma_align_kernel(
        const float* __restrict__ scores, const float* __restrict__ dec,
        const float* __restrict__ anchors, const int* __restrict__ labels,
        const float* __restrict__ gtb, const unsigned char* __restrict__ gmask,
        const float* __restrict__ atan_dec, const float* __restrict__ atan_gt,
        float* __restrict__ alignbuf) {
    int wave = threadIdx.x >> 5;
    int ln   = threadIdx.x & 31;
    int t = blockIdx.x * 8 + wave;          // grid sized exactly: NTILES/8 blocks
    int at = t % ATILES;
    int rem = t / ATILES;
    int gt = rem % GTILES;
    int b  = rem / GTILES;
    int a0 = at * 16, g0 = gt * 16;
    int n  = ln & 15;        // A-frag row M / B-frag col N / D col N
    int hi = ln >> 4;        // lane-half selects K pair (in) and M half (out)

    int a = a0 + n;
    int lbl = labels[b * GG + g0 + n];      // one-hot row for M = n
    const float* srow = scores + ((size_t)(b * AA + a)) * CC;
    __builtin_prefetch(srow, 0, 1);         // global_prefetch_b8

    v8f acc = {0.f, 0.f, 0.f, 0.f, 0.f, 0.f, 0.f, 0.f};
    #pragma unroll
    for (int k0 = 0; k0 < CC; k0 += 4) {
        int c = k0 + 2 * hi;
        v2f afrag;
        afrag.x = (lbl == c)     ? 1.0f : 0.0f;
        afrag.y = (lbl == c + 1) ? 1.0f : 0.0f;
        v2f bfrag = *(const v2f*)(srow + c);
        // D = A(16x4) x B(4x16) + C, exact in f32 (one-hot operand)
        acc = __builtin_amdgcn_wmma_f32_16x16x4_f32(
                false, afrag, false, bfrag, (short)0, acc, false, false);
    }

    // epilogue: lane owns column a, rows g = g0 + hi*8 + r
    float4 dbox = *(const float4*)(dec + ((size_t)(b * AA + a)) * 4);
    float ax = anchors[a * 2], ay = anchors[a * 2 + 1];
    float atd = atan_dec[b * AA + a];
    #pragma unroll
    for (int r = 0; r < 8; ++r) {
        int g = g0 + hi * 8 + r;
        float4 gbox = *(const float4*)(gtb + ((size_t)(b * GG + g)) * 4);
        float ci = ciou_f(gbox, dbox, atan_gt[b * GG + g], atd);
        float tpos = fmaxf(ci, 0.0f);
        float t2 = tpos * tpos;
        float t6 = t2 * t2 * t2;                       // relu(ciou)^6
        float al = sqrtf(fmaxf(acc[r], 0.0f)) * t6;    // score^0.5 * ciou^6
        bool inside = (gbox.x < ax) && (gbox.y < ay) && (gbox.z > ax) && (gbox.w > ay);
        if (!gmask[b * GG + g] || !inside) al = 0.0f;
        alignbuf[((size_t)(b * GG + g)) * AA + a] = al;
    }
}

// ---------------- 3: per-(b,g) top-10 + atomic scatter ----------------
// The align row (33.6 KB) is staged into LDS with the gfx1250 async
// memory->LDS engine (bypasses VGPRs, tracked by ASYNCcnt).
__global__ __launch_bounds__(256) void tal_topk_kernel(
        const float* __restrict__ alignbuf, const float* __restrict__ dec,
        const float* __restrict__ gtb, const float* __restrict__ atan_dec,
        const float* __restrict__ atan_gt,
        unsigned long long* __restrict__ best, float* __restrict__ norm) {
    __shared__ float s_al[AA];
    __shared__ float rv[256];
    __shared__ int   ri[256];
    __shared__ float cv[KK];
    __shared__ int   cidx[KK];

    int tid = threadIdx.x;
    int bg = blockIdx.x;
    int b = bg / GG, g = bg % GG;
    const float* row = alignbuf + (size_t)bg * AA;

    // async copy: global row -> LDS (GLOBAL_LOAD_ASYNC_TO_LDS_B32, GVS mode)
    unsigned lds_base = (unsigned)(size_t)(&s_al[0]);
    for (int i = tid; i < AA; i += 256) {
        unsigned lds_addr = lds_base + (unsigned)i * 4u;
        unsigned goff     = (unsigned)i * 4u;
        asm volatile("global_load_async_to_lds_b32 %0, %1, %2"
                     :: "v"(lds_addr), "v"(goff), "s"(row)
                     : "memory");
    }
#if __has_builtin(__builtin_amdgcn_s_wait_asynccnt)
    __builtin_amdgcn_s_wait_asynccnt(0);
#else
    asm volatile("s_wait_asynccnt 0" ::: "memory");
#endif
    __syncthreads();

    for (int k = 0; k < KK; ++k) {
        float bv = -1.0f; int bi = 0x7fffffff;
        for (int i = tid; i < AA; i += 256) {
            float v = s_al[i];
            if (v > bv || (v == bv && i < bi)) { bv = v; bi = i; }
        }
        rv[tid] = bv; ri[tid] = bi;
        __syncthreads();
        for (int off = 128; off > 0; off >>= 1) {
            if (tid < off) {
                float ov = rv[tid + off]; int oi = ri[tid + off];
                if (ov > rv[tid] || (ov == rv[tid] && oi < ri[tid])) {
                    rv[tid] = ov; ri[tid] = oi;
                }
            }
            __syncthreads();
        }
        if (tid == 0) {
            cv[k] = rv[0]; cidx[k] = ri[0];
            if (rv[0] > 0.0f) s_al[ri[0]] = -1.0f;   // mask out; align >= 0
        }
        __syncthreads();
    }

    if (tid == 0) {
        float4 gbox = *(const float4*)(gtb + ((size_t)(b * GG + g)) * 4);
        float atg = atan_gt[b * GG + g];
        float cio[KK];
        float max_align = 0.0f, max_ovl = 0.0f;
        for (int k = 0; k < KK; ++k) {
            if (cv[k] > 0.0f) {
                int a = cidx[k];
                float4 dbox = *(const float4*)(dec + ((size_t)(b * AA + a)) * 4);
                cio[k] = ciou_f(gbox, dbox, atg, atan_dec[b * AA + a]); // > 0 at matches
                max_align = fmaxf(max_align, cv[k]);
                max_ovl   = fmaxf(max_ovl, cio[k]);
            }
        }
        for (int k = 0; k < KK; ++k) {
            if (cv[k] > 0.0f) {
                int a = cidx[k];
                float nv = cv[k] * max_ovl / (max_align + EPSF);       // >= 0
                atomicMax((unsigned int*)&norm[b * AA + a], __float_as_uint(nv));
                // argmax over g of matched ciou, ties -> lowest g
                unsigned long long key =
                    ((unsigned long long)__float_as_uint(cio[k]) << 32) |
                    (unsigned long long)(unsigned)(GG - 1 - g);
                atomicMax(&best[b * AA + a], key);
            }
        }
    }
}

// ---------------- 4: cls one-hot * norm (coalesced over C) ----------------
__global__ void tal_cls_kernel(const unsigned long long* __restrict__ best,
                               const float* __restrict__ norm,
                               const int* __restrict__ labels,
                               float* __restrict__ out_cls) {
    int i = blockIdx.x * blockDim.x + threadIdx.x;
    if (i >= BB * AA * CC) return;
    int c  = i % CC;
    int ba = i / CC;
    unsigned long long key = best[ba];
    float o = 0.0f;
    if (key != 0ull) {
        int b = ba / AA;
        int g = (GG - 1) - (int)(key & 0xffffffffull);
        if (labels[b * GG + g] == c) o = norm[ba];
    }
    out_cls[i] = o;
}

// ---------------- 5: bbox / dist / fg ----------------
__global__ void tal_misc_kernel(const unsigned long long* __restrict__ best,
                                const float* __restrict__ norm,
                                const float* __restrict__ gtb,
                                const float* __restrict__ gdist,
                                float* __restrict__ out_bbox,
                                float* __restrict__ out_dist,
                                float* __restrict__ out_fg) {
    int ba = blockIdx.x * blockDim.x + threadIdx.x;
    if (ba >= BB * AA) return;
    unsigned long long key = best[ba];
    float4 bx = make_float4(-1.0f, -1.0f, -1.0f, -1.0f);
    float dd = -1.0f;
    if (key != 0ull) {
        int b = ba / AA;
        int g = (GG - 1) - (int)(key & 0xffffffffull);
        bx = *(const float4*)(gtb + ((size_t)(b * GG + g)) * 4);
        dd = gdist[b * GG + g];
    }
    *(float4*)(out_bbox + (size_t)ba * 4) = bx;
    out_dist[ba] = dd * norm[ba];   // unmatched: -1 * 0 (norm==0 there)
    out_fg[ba]   = 1.0f;            // argmax >= 0 always in reference
}

// ---------------- launcher ----------------
extern "C" void kernel_launch(void* const* d_in, const int* in_sizes, int n_in,
                              void* d_out, int out_size, void* d_ws, size_t ws_size,
                              hipStream_t stream) {
    const float* scores  = (const float*)d_in[0];         // [B,A,C]
    const float* dec     = (const float*)d_in[1];         // [B,A,4]
    // d_in[2] distances: unused by reference output
    const float* anchors = (const float*)d_in[3];         // [A,2]
    const int*   labels  = (const int*)d_in[4];           // [B,G]
    const float* gtb     = (const float*)d_in[5];         // [B,G,4]
    const float* gdist   = (const float*)d_in[6];         // [B,G,1]
    const unsigned char* gmask = (const unsigned char*)d_in[7]; // [B,G,1] bool

    // workspace layout (best first for 8-byte alignment)
    char* ws = (char*)d_ws;
    unsigned long long* best = (unsigned long long*)ws;                 // B*A u64
    float* norm     = (float*)(ws + (size_t)BB * AA * 8);               // B*A
    float* alignbuf = (float*)(ws + (size_t)BB * AA * 12);              // B*G*A
    float* atan_dec = alignbuf + (size_t)BB * GG * AA;                  // B*A
    float* atan_gt  = atan_dec + (size_t)BB * AA;                       // B*G

    float* out      = (float*)d_out;
    float* out_bbox = out;                                  // [B,A,4]
    float* out_cls  = out_bbox + (size_t)BB * AA * 4;       // [B,A,C]
    float* out_dist = out_cls  + (size_t)BB * AA * CC;      // [B,A,1]
    float* out_fg   = out_dist + (size_t)BB * AA;           // [B,A]

    tal_init_kernel<<<(BB * AA + 255) / 256, 256, 0, stream>>>(best, norm);
    tal_atan_kernel<<<(BB * AA + BB * GG + 255) / 256, 256, 0, stream>>>(
        dec, gtb, atan_dec, atan_gt);
    tal_wmma_align_kernel<<<NTILES / 8, 256, 0, stream>>>(
        scores, dec, anchors, labels, gtb, gmask, atan_dec, atan_gt, alignbuf);
    tal_topk_kernel<<<BB * GG, 256, 0, stream>>>(
        alignbuf, dec, gtb, atan_dec, atan_gt, best, norm);
    tal_cls_kernel<<<(BB * AA * CC + 255) / 256, 256, 0, stream>>>(
        best, norm, labels, out_cls);
    tal_misc_kernel<<<(BB * AA + 255) / 256, 256, 0, stream>>>(
        best, norm, gtb, gdist, out_bbox, out_dist, out_fg);
}